// loss_point_onlyNeg_27066883899872
// MI455X (gfx1250) — compile-verified
//
#include <hip/hip_runtime.h>
#include <math.h>

typedef __attribute__((ext_vector_type(2))) float v2f;
typedef __attribute__((ext_vector_type(8))) float v8f;

#define BDIM  256
#define NCTR  16      // centers per block (16 | 96, so one image per block)
#define NPTS  1080    // L*S = 6*180 gt points per image
#define FPAD  257     // padded feat row (16-row stride conflict-free in LDS)

__global__ __launch_bounds__(BDIM)
void center_block_kernel(const float* __restrict__ lane_xs,
                         const float* __restrict__ lane_ys,
                         const float* __restrict__ centers,
                         const float* __restrict__ feat,   // (8,256,90,160)
                         const float* __restrict__ Wn,     // (256,2)
                         const float* __restrict__ bn,     // (2,)
                         float* __restrict__ partials)     // (48,2)
{
    __shared__ float gtx[NPTS];
    __shared__ float gty[NPTS];
    __shared__ float featLDS[NCTR * FPAD];   // 16 x 256 feature tile (padded)
    __shared__ float WpadLDS[256 * 16];      // W zero-padded to 16 columns
    __shared__ float cxs[NCTR], cys[NCTR];
    __shared__ int   spat[NCTR];
    __shared__ float selLDS[NCTR];
    __shared__ float ceLDS[NCTR];
    __shared__ float minred[BDIM];
    __shared__ float logitsLDS[NCTR * 2];

    const int tid = threadIdx.x;
    const int blk = blockIdx.x;
    const int n0  = blk * NCTR;       // first center of this block
    const int b   = blk / 6;          // image index (96 centers per image)

    // ---- Phase 0: per-center setup + stage zero-padded W panel ----
    if (tid < NCTR) {
        float cx = centers[(n0 + tid) * 2 + 0] * 1280.0f;
        float cy = centers[(n0 + tid) * 2 + 1] * 720.0f;
        cxs[tid] = cx; cys[tid] = cy;
        float x1 = cx - 16.0f, y1 = cy - 16.0f;
        float x2 = cx + 16.0f, y2 = cy + 16.0f;
        bool valid = (x1 > 0.0f) && (x1 < 1280.0f) &&
                     (y1 >= 0.0f) && (y1 < 720.0f) &&
                     (x2 >= 0.0f) && (x2 < 1280.0f) &&
                     (y2 >= 0.0f) && (y2 < 720.0f);
        selLDS[tid] = valid ? 1.0f : 0.0f;
        int fx = (int)(cx * 0.125f); fx = fx < 0 ? 0 : (fx > 159 ? 159 : fx);
        int fy = (int)(cy * 0.125f); fy = fy < 0 ? 0 : (fy > 89  ? 89  : fy);
        spat[tid] = fy * 160 + fx;
    }
    for (int i = tid; i < 256 * 16; i += BDIM) {
        int k = i >> 4, n = i & 15;
        WpadLDS[i] = (n < 2) ? Wn[k * 2 + n] : 0.0f;
    }

    // ---- Phase 1: build this image's gt points (linear resize 72 -> 180) ----
    for (int j = tid; j < NPTS; j += BDIM) {
        int l = j / 180, s = j % 180;
        int lane = b * 6 + l;
        float src = (float)s * (71.0f / 179.0f);
        int i0 = (int)floorf(src); if (i0 > 71) i0 = 71;
        int i1 = i0 + 1;           if (i1 > 71) i1 = 71;
        float w = src - (float)i0;
        gtx[j] = (lane_xs[lane * 72 + i0] * (1.0f - w) +
                  lane_xs[lane * 72 + i1] * w) * 1280.0f;
        gty[j] = (lane_ys[lane * 72 + i0] * (1.0f - w) +
                  lane_ys[lane * 72 + i1] * w) * 720.0f;
    }
    __syncthreads();

    // ---- Phase 2: min squared distance, 16 partial scanners per center ----
    {
        int c = tid & 15, g = tid >> 4;
        float cx = cxs[c], cy = cys[c];
        float mn = 3.4e38f;
        for (int j = g; j < NPTS; j += 16) {
            float dx = cx - gtx[j];
            float dy = cy - gty[j];
            mn = fminf(mn, dx * dx + dy * dy);
        }
        minred[tid] = mn;
    }
    __syncthreads();
    for (int off = 128; off >= 16; off >>= 1) {
        if (tid < off) minred[tid] = fminf(minred[tid], minred[tid + off]);
        __syncthreads();
    }
    if (tid < NCTR) {
        float md = sqrtf(minred[tid]);
        bool ok = (md > 22.627416997969522f) && (md < 130.0f); // sqrt(2)*16, 130
        selLDS[tid] = ok ? selLDS[tid] : 0.0f;
    }

    // ---- Phase 3: gather 16x256 feature tile into LDS ----
    {
        long imgbase = (long)b * 256 * 14400;   // 14400 = 90*160
        for (int idx = tid; idx < NCTR * 256; idx += BDIM) {
            int cc = idx >> 8, ch = idx & 255;
            featLDS[cc * FPAD + ch] = feat[imgbase + (long)ch * 14400 + spat[cc]];
        }
    }
    __syncthreads();

    // ---- Phase 4: logits = feat @ W + b via V_WMMA_F32_16X16X4_F32 ----
    // All 8 waves run this uniformly (EXEC all-1s, as WMMA requires).
    // Unconditional LDS loads only: no exec-masked predication in the loop.
    {
        int lane = tid & 31;
        int nIdx = lane & 15;     // N column for B/C fragments
        int half = lane >> 4;
        int m    = lane & 15;     // M row for A fragment
        v8f cacc;
        float cinit = (nIdx < 2) ? bn[nIdx] : 0.0f;   // fold bias into C
        for (int v = 0; v < 8; ++v) cacc[v] = cinit;

        const float* aBase = &featLDS[m * FPAD + 2 * half];
        const float* bBase = &WpadLDS[2 * half * 16 + nIdx];
        for (int kc = 0; kc < 64; ++kc) {
            v2f a, bb;
            a.x  = aBase[kc * 4];
            a.y  = aBase[kc * 4 + 1];
            bb.x = bBase[kc * 64];        // row kA   (16 floats/row)
            bb.y = bBase[kc * 64 + 16];   // row kA+1
            cacc = __builtin_amdgcn_wmma_f32_16x16x4_f32(
                false, a, false, bb, (short)0, cacc, false, false);
        }
        if (tid < 32 && nIdx < 2) {
            for (int v = 0; v < 8; ++v)
                logitsLDS[(v + 8 * half) * 2 + nIdx] = cacc[v];
        }
    }
    __syncthreads();

    // ---- Phase 5: cross-entropy, masked partial sums ----
    if (tid < NCTR) {
        float l0 = logitsLDS[tid * 2 + 0];
        float l1 = logitsLDS[tid * 2 + 1];
        float mx = fmaxf(l0, l1);
        float ce = logf(expf(l0 - mx) + expf(l1 - mx)) + mx - l0;
        ceLDS[tid] = ce * selLDS[tid];
    }
    __syncthreads();
    if (tid == 0) {
        float sce = 0.0f, ssel = 0.0f;
        for (int i = 0; i < NCTR; ++i) { sce += ceLDS[i]; ssel += selLDS[i]; }
        partials[blk * 2 + 0] = sce;
        partials[blk * 2 + 1] = ssel;
    }
}

__global__ void finalize_kernel(const float* __restrict__ partials,
                                float* __restrict__ out)
{
    if (threadIdx.x == 0) {
        float sce = 0.0f, ssel = 0.0f;
        for (int i = 0; i < 48; ++i) {
            sce  += partials[i * 2 + 0];
            ssel += partials[i * 2 + 1];
        }
        out[0] = sce / fmaxf(ssel, 1.0f);
    }
}

extern "C" void kernel_launch(void* const* d_in, const int* in_sizes, int n_in,
                              void* d_out, int out_size, void* d_ws, size_t ws_size,
                              hipStream_t stream)
{
    const float* lane_xs = (const float*)d_in[0];  // (48,72)
    const float* lane_ys = (const float*)d_in[1];  // (48,72)
    const float* centers = (const float*)d_in[2];  // (768,2)
    const float* feat    = (const float*)d_in[3];  // (8,256,90,160)
    const float* Wn      = (const float*)d_in[4];  // (256,2)
    const float* bn      = (const float*)d_in[5];  // (2,)
    float* ws  = (float*)d_ws;                     // 96 floats of partials
    float* out = (float*)d_out;

    center_block_kernel<<<48, BDIM, 0, stream>>>(lane_xs, lane_ys, centers,
                                                 feat, Wn, bn, ws);
    finalize_kernel<<<1, 32, 0, stream>>>(ws, out);
}